// DrugClassificationModel_9672266351385
// MI455X (gfx1250) — compile-verified
//
#include <hip/hip_runtime.h>
#include <math.h>

typedef float v2f __attribute__((ext_vector_type(2)));
typedef float v8f __attribute__((ext_vector_type(8)));

#define NNODES 40000
#define NEDGES 640000
#define NGRAPH 256
#define IN_F   128
#define C1     128
#define C2     256
#define C3     512
#define CLH    1024
#define NLAB   12

// ---------------- small elementwise / scatter kernels ----------------

__global__ void fill_f(float* p, int n, float v) {
  int i = blockIdx.x * blockDim.x + threadIdx.x;
  if (i < n) p[i] = v;
}

__global__ void deg_edges_k(const int* __restrict__ dst, float* deg, int e) {
  int i = blockIdx.x * blockDim.x + threadIdx.x;
  if (i < e) atomicAdd(&deg[dst[i]], 1.0f);
}

__global__ void rsqrt_k(const float* __restrict__ deg, float* dis, int n) {
  int i = blockIdx.x * blockDim.x + threadIdx.x;
  if (i < n) dis[i] = rsqrtf(deg[i]);
}

__global__ void norm_k(const int* __restrict__ src, const int* __restrict__ dst,
                       const float* __restrict__ dis, float* nrm, int e) {
  int i = blockIdx.x * blockDim.x + threadIdx.x;
  if (i < e) nrm[i] = dis[src[i]] * dis[dst[i]];
}

// s0[n] = dot(v[n,:], W0[:,0]); one wave per node, coalesced over k
__global__ void dot_w0_k(const float* __restrict__ v, const float* __restrict__ w0,
                         float* s, int n) {
  int wave = blockIdx.x * (blockDim.x >> 5) + (threadIdx.x >> 5);
  int lane = threadIdx.x & 31;
  if (wave >= n) return;
  const float* row = v + (size_t)wave * IN_F;
  float acc = row[lane] * w0[lane] + row[lane + 32] * w0[lane + 32] +
              row[lane + 64] * w0[lane + 64] + row[lane + 96] * w0[lane + 96];
  #pragma unroll
  for (int off = 16; off > 0; off >>= 1) acc += __shfl_down(acc, off, 32);
  if (lane == 0) s[wave] = acc;
}

// out[n] = dis[n]^2 * x[n] + bias   (self-loop term doubles as init)
__global__ void sprop_init_k(const float* __restrict__ dis, const float* __restrict__ x,
                             const float* bias, float* out, int n) {
  int i = blockIdx.x * blockDim.x + threadIdx.x;
  if (i < n) out[i] = dis[i] * dis[i] * x[i] + (bias ? bias[0] : 0.0f);
}

__global__ void sprop_edge_k(const int* __restrict__ src, const int* __restrict__ dst,
                             const float* __restrict__ nrm, const float* __restrict__ x,
                             float* out, int e) {
  int i = blockIdx.x * blockDim.x + threadIdx.x;
  if (i < e) atomicAdd(&out[dst[i]], nrm[i] * x[src[i]]);
}

// x1[n,c] = relu(p1[n]*W1[c] + b1[c])   (rank-1 expand of layer 1)
__global__ void expand1_k(const float* __restrict__ p1, const float* __restrict__ w1,
                          const float* __restrict__ b1, float* x1, int total) {
  int i = blockIdx.x * blockDim.x + threadIdx.x;
  if (i < total) {
    int node = i >> 7, c = i & 127;
    x1[i] = fmaxf(fmaf(p1[node], w1[c], b1[c]), 0.0f);
  }
}

__global__ void vprop_init_k(const float* __restrict__ dis, const float* __restrict__ x,
                             float* out, int logC, int total) {
  int i = blockIdx.x * blockDim.x + threadIdx.x;
  if (i < total) {
    float d = dis[i >> logC];
    out[i] = d * d * x[i];
  }
}

__global__ void vprop_edge_k(const int* __restrict__ src, const int* __restrict__ dst,
                             const float* __restrict__ nrm, const float* __restrict__ x,
                             float* out, int logC, int cmask, long total) {
  long i = (long)blockIdx.x * blockDim.x + threadIdx.x;
  if (i < total) {
    int e = (int)(i >> logC);
    int c = (int)i & cmask;
    atomicAdd(&out[((size_t)dst[e] << logC) + c], nrm[e] * x[((size_t)src[e] << logC) + c]);
  }
}

// ---------------- fp32 WMMA GEMM: C = act(A @ B + bias) ----------------
// One wave computes a 16 x (16*NT) tile via V_WMMA_F32_16X16X4_F32.
// Fragment layout per CDNA5 ISA 7.12.2:
//  A (16x4):  lane%16 = M row; VGPR0 = K = 2*(lane/16), VGPR1 = K+1
//  B (4x16):  lane%16 = N col; VGPR0 = K = 2*(lane/16), VGPR1 = K+1
//  C (16x16): lane%16 = N col; VGPR r = M row (r + 8*(lane/16))
template <int NT>
__global__ __launch_bounds__(32) void gemm_wmma_k(
    const float* __restrict__ A, const float* __restrict__ B,
    const float* __restrict__ bias, float* __restrict__ C,
    int M, int K, int N, int doRelu) {
  const int lane = threadIdx.x & 31;
  const int hf   = lane >> 4;
  const int l16  = lane & 15;
  const int row  = blockIdx.y * 16 + l16;
  const int ncol0 = blockIdx.x * (16 * NT);

  v8f acc[NT] = {};

  for (int k0 = 0; k0 < K; k0 += 4) {
    int ka = k0 + 2 * hf;
    v2f a;
    a.x = A[(size_t)row * K + ka];
    a.y = A[(size_t)row * K + ka + 1];
    #pragma unroll
    for (int t = 0; t < NT; ++t) {
      int n = ncol0 + t * 16 + l16;
      v2f b;
      b.x = B[(size_t)ka * N + n];
      b.y = B[(size_t)(ka + 1) * N + n];
      acc[t] = __builtin_amdgcn_wmma_f32_16x16x4_f32(
          false, a, false, b, (short)0, acc[t], false, false);
    }
  }

  #pragma unroll
  for (int t = 0; t < NT; ++t) {
    int n = ncol0 + t * 16 + l16;
    float bv = bias ? bias[n] : 0.0f;
    #pragma unroll
    for (int r = 0; r < 8; ++r) {
      int m = blockIdx.y * 16 + hf * 8 + r;
      float val = acc[t][r] + bv;
      if (doRelu) val = fmaxf(val, 0.0f);
      C[(size_t)m * N + n] = val;
    }
  }
}

// ---------------- pooling ----------------

__global__ void count_k(const int* __restrict__ batch, float* counts, int n) {
  int i = blockIdx.x * blockDim.x + threadIdx.x;
  if (i < n) atomicAdd(&counts[batch[i]], 1.0f);
}

__global__ void pool_scatter_k(const int* __restrict__ batch, const float* __restrict__ x,
                               float* sums, int* mx, int total) {
  int i = blockIdx.x * blockDim.x + threadIdx.x;
  if (i < total) {
    int node = i >> 9, c = i & 511;
    int b = batch[node];
    float v = x[i];  // post-relu: v >= 0, so int bit-pattern order == float order
    atomicAdd(&sums[b * C3 + c], v);
    atomicMax(&mx[b * C3 + c], __float_as_int(v));
  }
}

__global__ void pool_final_k(const float* __restrict__ sums, const int* __restrict__ mx,
                             const float* __restrict__ counts, float* g, int total) {
  int i = blockIdx.x * blockDim.x + threadIdx.x;
  if (i < total) {
    int b = i >> 9;
    g[i] = sums[i] / fmaxf(counts[b], 1.0f) + __int_as_float(mx[i]);
  }
}

// ---------------- BatchNorm over graphs (axis 0), in place ----------------
__global__ void bn_k(float* gl, const float* __restrict__ w, const float* __restrict__ b) {
  int c = blockIdx.x;          // 0..CLH-1
  int t = threadIdx.x;         // 0..NGRAPH-1
  __shared__ float red[NGRAPH];
  float v = gl[(size_t)t * CLH + c];
  red[t] = v;
  __syncthreads();
  for (int s = NGRAPH / 2; s > 0; s >>= 1) {
    if (t < s) red[t] += red[t + s];
    __syncthreads();
  }
  float mu = red[0] * (1.0f / NGRAPH);
  __syncthreads();
  float d = v - mu;
  red[t] = d * d;
  __syncthreads();
  for (int s = NGRAPH / 2; s > 0; s >>= 1) {
    if (t < s) red[t] += red[t + s];
    __syncthreads();
  }
  float var = red[0] * (1.0f / NGRAPH);
  gl[(size_t)t * CLH + c] = d * rsqrtf(var + 1e-5f) * w[c] + b[c];
}

// ---------------- classifier + sigmoid ----------------
__global__ void classifier_k(const float* __restrict__ gl, const float* __restrict__ Wc,
                             const float* __restrict__ bc, float* out) {
  int i = blockIdx.x * blockDim.x + threadIdx.x;
  if (i >= NGRAPH * NLAB) return;
  int g = i / NLAB, l = i % NLAB;
  float acc = bc[l];
  const float* row = gl + (size_t)g * CLH;
  for (int k = 0; k < CLH; ++k) acc = fmaf(row[k], Wc[k * NLAB + l], acc);
  out[i] = acc;
  out[NGRAPH * NLAB + i] = 1.0f / (1.0f + expf(-acc));
}

// ---------------- launch ----------------

extern "C" void kernel_launch(void* const* d_in, const int* in_sizes, int n_in,
                              void* d_out, int out_size, void* d_ws, size_t ws_size,
                              hipStream_t stream) {
  (void)in_sizes; (void)n_in; (void)out_size; (void)ws_size;
  const float* v    = (const float*)d_in[0];
  const int*   edges= (const int*)d_in[1];
  const int*   batch= (const int*)d_in[2];
  const float* W0 = (const float*)d_in[3];
  const float* b0 = (const float*)d_in[4];
  const float* W1 = (const float*)d_in[5];
  const float* b1 = (const float*)d_in[6];
  const float* W2 = (const float*)d_in[7];
  const float* b2 = (const float*)d_in[8];
  const float* W3 = (const float*)d_in[9];
  const float* b3 = (const float*)d_in[10];
  const float* Wl = (const float*)d_in[11];
  const float* bl = (const float*)d_in[12];
  const float* bnw= (const float*)d_in[13];
  const float* bnb= (const float*)d_in[14];
  const float* Wc = (const float*)d_in[15];
  const float* bc = (const float*)d_in[16];

  const int* src = edges;
  const int* dst = edges + NEDGES;

  float* w = (float*)d_ws;
  size_t off = 0;
  auto alloc = [&](size_t n) { float* p = w + off; off += n; return p; };
  float* deg    = alloc(NNODES);          // later reused as p1
  float* dis    = alloc(NNODES);
  float* nrm    = alloc(NEDGES);
  float* sc0    = alloc(NNODES);          // s0 = v@W0
  float* sc1    = alloc(NNODES);          // x0 = P s0 + b0
  float* counts = alloc(NGRAPH);
  float* sums   = alloc((size_t)NGRAPH * C3);
  float* mxf    = alloc((size_t)NGRAPH * C3);
  float* g      = alloc((size_t)NGRAPH * C3);
  float* gl     = alloc((size_t)NGRAPH * CLH);
  float* bufB   = alloc((size_t)NNODES * C2);   // q2 / q3
  float* bufA   = alloc((size_t)NNODES * C3);   // x1 / x2 / x3

  const int T = 256;
  const int gN  = (NNODES + T - 1) / T;
  const int gE  = NEDGES / T;                   // 2500

  // degrees (init 1.0 = self loop), dis = deg^-1/2, edge norms
  fill_f<<<gN, T, 0, stream>>>(deg, NNODES, 1.0f);
  deg_edges_k<<<gE, T, 0, stream>>>(dst, deg, NEDGES);
  rsqrt_k<<<gN, T, 0, stream>>>(deg, dis, NNODES);
  norm_k<<<gE, T, 0, stream>>>(src, dst, dis, nrm, NEDGES);

  // layer 0 (out dim 1): s0 = v@W0 ; x0 = P s0 + b0
  dot_w0_k<<<NNODES / 8, T, 0, stream>>>(v, W0, sc0, NNODES);
  sprop_init_k<<<gN, T, 0, stream>>>(dis, sc0, b0, sc1, NNODES);
  sprop_edge_k<<<gE, T, 0, stream>>>(src, dst, nrm, sc0, sc1, NEDGES);

  // layer 1 rank-1: p1 = P x0 ; x1 = relu(p1 ⊗ W1 + b1)
  float* p1 = deg;  // deg no longer needed
  sprop_init_k<<<gN, T, 0, stream>>>(dis, sc1, nullptr, p1, NNODES);
  sprop_edge_k<<<gE, T, 0, stream>>>(src, dst, nrm, sc1, p1, NEDGES);
  float* x1 = bufA;
  expand1_k<<<(NNODES * C1) / T, T, 0, stream>>>(p1, W1, b1, x1, NNODES * C1);

  // layer 2: q2 = P x1 (128 ch) ; x2 = relu(q2 @ W2 + b2)
  float* q2 = bufB;
  vprop_init_k<<<(NNODES * C1) / T, T, 0, stream>>>(dis, x1, q2, 7, NNODES * C1);
  vprop_edge_k<<<((long)NEDGES * C1) / T, T, 0, stream>>>(src, dst, nrm, x1, q2, 7, 127,
                                                          (long)NEDGES * C1);
  float* x2 = bufA;
  gemm_wmma_k<4><<<dim3(C2 / 64, NNODES / 16), 32, 0, stream>>>(q2, W2, b2, x2,
                                                                NNODES, C1, C2, 1);

  // layer 3: q3 = P x2 (256 ch) ; x3 = relu(q3 @ W3 + b3)
  float* q3 = bufB;
  vprop_init_k<<<(NNODES * C2) / T, T, 0, stream>>>(dis, x2, q3, 8, NNODES * C2);
  vprop_edge_k<<<((long)NEDGES * C2) / T, T, 0, stream>>>(src, dst, nrm, x2, q3, 8, 255,
                                                          (long)NEDGES * C2);
  float* x3 = bufA;
  gemm_wmma_k<4><<<dim3(C3 / 64, NNODES / 16), 32, 0, stream>>>(q3, W3, b3, x3,
                                                                NNODES, C2, C3, 1);

  // pooling: mean + max per graph
  fill_f<<<1, T, 0, stream>>>(counts, NGRAPH, 0.0f);
  fill_f<<<(NGRAPH * C3) / T, T, 0, stream>>>(sums, NGRAPH * C3, 0.0f);
  fill_f<<<(NGRAPH * C3) / T, T, 0, stream>>>(mxf, NGRAPH * C3, 0.0f);
  count_k<<<gN, T, 0, stream>>>(batch, counts, NNODES);
  pool_scatter_k<<<(NNODES * C3) / T, T, 0, stream>>>(batch, x3, sums, (int*)mxf,
                                                      NNODES * C3);
  pool_final_k<<<(NGRAPH * C3) / T, T, 0, stream>>>(sums, (const int*)mxf, counts, g,
                                                    NGRAPH * C3);

  // head: gl = g @ Wl + bl ; BN ; logits = gl' @ Wc + bc ; sigmoid
  gemm_wmma_k<4><<<dim3(CLH / 64, NGRAPH / 16), 32, 0, stream>>>(g, Wl, bl, gl,
                                                                 NGRAPH, C3, CLH, 0);
  bn_k<<<CLH, NGRAPH, 0, stream>>>(gl, bnw, bnb);
  classifier_k<<<(NGRAPH * NLAB + T - 1) / T, T, 0, stream>>>(gl, Wc, bc, (float*)d_out);
}